// Sineconv_33724083208583
// MI455X (gfx1250) — compile-verified
//
#include <hip/hip_runtime.h>

// Sineconv for MI455X (gfx1250).
// out[b,l,c] = sum_{a<8,q<33} x[b, l+((33a+q)>>3), c] * sines[b, l+q, 8c+a]
//            + x[b, l+16, c] * W_res[c]
// B=4, L=8192, CIN=8, lout=8160.  ~138 MFLOP, ~2MB HBM -> latency bound.
// CDNA5 path: TDM async tensor loads (TENSORcnt) stage x / sine_range tiles
// into LDS, overlapped with building the sine table; VALU FMA main loop with
// fully constant-folded DS offsets and conflict-free bank mapping.

#define TILE 64
#define HALO 32
#define TT (TILE + HALO) /* 96 rows per tile */
#define SSTR 66          /* padded sine row stride (floats): bank = 2g+8c+a */
#define LGRID 128        /* tiles per batch: ceil(8160/64) */
#define LOUT 8160
#define LSEQ 8192
#define NB 4
#define NCIN 8
#define TWO_PI 6.28318530717958647692f

typedef unsigned int v4u __attribute__((ext_vector_type(4)));
typedef int v8i __attribute__((ext_vector_type(8)));
typedef int v4i __attribute__((ext_vector_type(4)));

#if defined(__has_builtin)
#if __has_builtin(__builtin_amdgcn_tensor_load_to_lds) && \
    __has_builtin(__builtin_amdgcn_s_wait_tensorcnt)
#define USE_TDM 1
#endif
#endif
#ifndef USE_TDM
#define USE_TDM 0
#endif

#if USE_TDM
// 1-D TDM descriptor (ISA cdna5 8.3/8.4): count=1, type=2 ("image"),
// data_size=4B, tile_dim1/2=0 (unused -> 1D), groups 2/3 zero (<=2D tensor),
// group 4 (VADDR4 slot) zero per ISA "unused / NULL".
// tensor_dim0 = remaining valid elements so OOB reads return zero (tail tile).
__device__ __forceinline__ void tdm_load_1d_f32(unsigned lds_byte_off,
                                                const float* gsrc,
                                                unsigned valid_elems,
                                                unsigned tile_elems) {
  unsigned long long ga = (unsigned long long)(const void*)gsrc;
  v4u g0;
  g0.x = 1u;                                    // count=1 (valid descriptor)
  g0.y = lds_byte_off;                          // lds_addr
  g0.z = (unsigned)(ga & 0xffffffffull);        // global_addr[31:0]
  g0.w = (unsigned)((ga >> 32) & 0x1ffffffull)  // global_addr[56:32]
         | (2u << 30);                          // type = 2
  v8i g1;
  g1[0] = (int)(2u << 16);                        // data_size=4B, no flags
  g1[1] = (int)((valid_elems & 0xffffu) << 16);   // tensor_dim0[15:0]
  g1[2] = (int)((valid_elems >> 16) & 0xffffu);   // tensor_dim0[31:16]
  g1[3] = (int)((tile_elems & 0xffffu) << 16);    // tile_dim0
  g1[4] = 0;                                      // tile_dim1=0, tile_dim2=0
  g1[5] = (int)valid_elems;                       // tensor_dim0_stride[31:0]
  g1[6] = 0;
  g1[7] = 0;
  v4i z4 = {0, 0, 0, 0};
  v8i z8 = {0, 0, 0, 0, 0, 0, 0, 0};
  __builtin_amdgcn_tensor_load_to_lds(g0, g1, z4, z4, z8, 0);
}
#endif

__global__ __launch_bounds__(256) void sineconv_kernel(
    const float* __restrict__ x, const float* __restrict__ sine_range,
    const float* __restrict__ phases, const float* __restrict__ amplitude,
    const float* __restrict__ frequency, const float* __restrict__ W_res,
    float* __restrict__ out) {
  __shared__ __align__(16) float sh_sr[TT];
  __shared__ __align__(16) float sh_x[TT * NCIN];
  __shared__ __align__(16) float sh_s[TT * SSTR];

  const int tid = threadIdx.x;
  const int b = blockIdx.x >> 7;  // / LGRID
  const int tileid = blockIdx.x & (LGRID - 1);
  const int l0 = tileid * TILE;
  const int rows_valid = LSEQ - l0;  // >= TILE always

  // ---- stage x / sine_range tiles into LDS ----
#if USE_TDM
  if (tid < 32) {  // wave 0 only issues the DMAs (EXEC ignored by TDM)
    tdm_load_1d_f32((unsigned)(size_t)(&sh_sr[0]),
                    sine_range + (size_t)b * LSEQ + l0, (unsigned)rows_valid,
                    TT);
    tdm_load_1d_f32((unsigned)(size_t)(&sh_x[0]),
                    x + ((size_t)b * LSEQ + l0) * NCIN,
                    (unsigned)rows_valid * NCIN, TT * NCIN);
    __builtin_amdgcn_s_wait_tensorcnt(1);  // sine_range tile landed (in-order)
  }
  __syncthreads();
#else
  for (int i = tid; i < TT; i += 256)
    sh_sr[i] = (i < rows_valid) ? sine_range[(size_t)b * LSEQ + l0 + i] : 0.0f;
  for (int i = tid; i < TT * NCIN; i += 256)
    sh_x[i] =
        (i < rows_valid * NCIN) ? x[((size_t)b * LSEQ + l0) * NCIN + i] : 0.0f;
  __syncthreads();
#endif

  // ---- build sine table sh_s[t][chan] (overlaps with x-tile DMA) ----
  {
    const int chan = tid & 63;
    const float coef = frequency[chan] * (19000.0f * TWO_PI);
    const float phv = phases[chan] * TWO_PI;
    const float ampv = amplitude[chan];
    for (int t = tid >> 6; t < TT; t += 4)
      sh_s[t * SSTR + chan] = ampv * __sinf(fmaf(coef, sh_sr[t], phv));
  }

#if USE_TDM
  if (tid < 32) __builtin_amdgcn_s_wait_tensorcnt(0);  // x tile landed
#endif
  __syncthreads();

  // ---- main correlation: 2 outputs per thread ----
  // lane map: c = lane>>2 (0..7), g = lane&3; wave w covers l = l0+8w .. +7.
  // sine bank = (2g + 8c + a) mod 64 -> conflict free; x bank = (8g + c).
  const int lane = tid & 31;
  const int wv = tid >> 5;
  const int g = lane & 3;
  const int c = lane >> 2;
  const float wres = W_res[c];
  const int ltb = wv * 8 + g;

#pragma unroll
  for (int j = 0; j < 2; ++j) {
    const int lt = ltb + 4 * j;
    const int l = l0 + lt;
    float acc = wres * sh_x[(lt + 16) * NCIN + c];  // residual term
    const float* xb = &sh_x[lt * NCIN + c];
    const float* sb = &sh_s[lt * SSTR + 8 * c];
#pragma unroll
    for (int q = 0; q < 33; ++q) {
#pragma unroll
      for (int a = 0; a < 8; ++a) {
        const int u = (33 * a + q) >> 3;  // compile-time constant
        acc = fmaf(xb[u * NCIN], sb[q * SSTR + a], acc);
      }
    }
    if (l < LOUT) out[((size_t)b * LOUT + l) * NCIN + c] = acc;
  }
}

extern "C" void kernel_launch(void* const* d_in, const int* in_sizes, int n_in,
                              void* d_out, int out_size, void* d_ws,
                              size_t ws_size, hipStream_t stream) {
  (void)in_sizes;
  (void)n_in;
  (void)out_size;
  (void)d_ws;
  (void)ws_size;
  const float* x = (const float*)d_in[0];
  const float* sine_range = (const float*)d_in[1];
  const float* phases = (const float*)d_in[2];
  const float* amplitude = (const float*)d_in[3];
  const float* frequency = (const float*)d_in[4];
  const float* W_res = (const float*)d_in[5];
  float* out = (float*)d_out;

  dim3 grid(NB * LGRID);  // 512 blocks
  dim3 block(256);        // 8 wave32s
  sineconv_kernel<<<grid, block, 0, stream>>>(x, sine_range, phases, amplitude,
                                              frequency, W_res, out);
}